// ODE_RNN_74208444940297
// MI455X (gfx1250) — compile-verified
//
#include <hip/hip_runtime.h>

// ---------------------------------------------------------------------------
// ODE-RNN (Tsit5 neural ODE between RNN updates) for MI455X / gfx1250.
// Strategy: lockstep-batch 16 sequences per wave32; every tiny matmul is a
// chain of V_WMMA_F32_16X16X4_F32 ops (weights = A operand preloaded in VGPRs,
// activations = B operand staged through a transposed 16x16 LDS tile).
// This revision: use CDNA5's hardware V_TANH_F32 when the builtin exists —
// the recurrence is critical-path bound, so collapsing each 9-op software
// tanh into one TRANS instruction directly shortens the dependent chain.
// ---------------------------------------------------------------------------

typedef __attribute__((ext_vector_type(2))) float v2f;
typedef __attribute__((ext_vector_type(4))) float v4f;
typedef __attribute__((ext_vector_type(8))) float v8f;

#define Bsz    256
#define Nseq   512
#define Iin    32
#define Hd     8
#define Wd     16
#define Od     8
#define NSUB   8
#define TILE   16   // sequences per wave (== WMMA N dimension)

// Tsit5 tableau
#define A21f 0.161f
#define A31f (-0.008480655492356989f)
#define A32f 0.335480655492357f
#define A41f 2.8971530571054935f
#define A42f (-6.359448489975075f)
#define A43f 4.3622954328695815f
#define A51f 5.325864828439257f
#define A52f (-11.748883564062828f)
#define A53f 7.4955393428898365f
#define A54f (-0.09249506636175525f)
#define A61f 5.86145544294642f
#define A62f (-12.92096931784711f)
#define A63f 8.159367898576159f
#define A64f (-0.071584973281401f)
#define A65f (-0.028269050394068383f)
#define B1f 0.09646076681806523f
#define B2f 0.01f
#define B3f 0.4798896504144996f
#define B4f 1.379008574103742f
#define B5f (-3.290069515436081f)
#define B6f 2.324710524099774f
#define C2f 0.161f
#define C3f 0.327f
#define C4f 0.9f
#define C5f 0.9800255409045097f

__device__ __forceinline__ float fast_tanh(float x) {
#if __has_builtin(__builtin_amdgcn_tanhf)
  return __builtin_amdgcn_tanhf(x);                 // v_tanh_f32 (1 TRANS op)
#else
  x = fminf(8.0f, fmaxf(-8.0f, x));
  float e = __expf(2.0f * x);                       // v_exp_f32 fallback
  return (e - 1.0f) * __builtin_amdgcn_rcpf(e + 1.0f);
#endif
}

__device__ __forceinline__ v8f wmma4(v2f a, v2f b, v8f c) {
  // D = A(16xK=4, f32) * B(4x16, f32) + C(16x16, f32)
  return __builtin_amdgcn_wmma_f32_16x16x4_f32(
      /*neg_a=*/false, a, /*neg_b=*/false, b,
      /*c_mod=*/(short)0, c, /*reuse_a=*/false, /*reuse_b=*/false);
}

// A-matrix K-slice: lane m (=lane&15) carries row m; lane-half h supplies
// K = {base+2h, base+2h+1} per the 16x4 f32 A layout. Zero-pads OOB.
__device__ __forceinline__ v2f aslice(const float* W, int rows, int cols, int m, int k) {
  v2f a;
  a.x = (m < rows && (k    ) < cols) ? W[m * cols + k    ] : 0.0f;
  a.y = (m < rows && (k + 1) < cols) ? W[m * cols + k + 1] : 0.0f;
  return a;
}

// C/D accumulator bias init: VGPR r holds row (r + 8h); bias is per-row.
__device__ __forceinline__ v8f cbias(const float* b, int len, int h) {
  v8f c;
#pragma unroll
  for (int r = 0; r < 8; ++r) {
    int i = r + 8 * h;
    c[r] = (i < len) ? b[i] : 0.0f;
  }
  return c;
}

struct VFW {   // vector-field MLP weights, WMMA-resident
  v2f aw0[3], aw1[4], aw2[4];
  v8f cb0, cb1, cb2;
  float scale;
};

// One vector-field eval: k = scale * tanh( tanh(W2 tanh(W1 tanh(W0 [t;y]))) )
// actT is a transposed 16x16 LDS tile: actT[col*16 + k].
__device__ __forceinline__ v8f vf_eval(const VFW& P, float* actT,
                                       int n, int kh, int h, bool lo,
                                       float t, const v8f y) {
  if (lo) {  // columns owned by lanes 0..15; write x = [t, y0..y7, 0,0,0]
    *(v4f*)(actT + n * 16 + 0) = (v4f){t,    y[0], y[1], y[2]};
    *(v4f*)(actT + n * 16 + 4) = (v4f){y[3], y[4], y[5], y[6]};
    *(v4f*)(actT + n * 16 + 8) = (v4f){y[7], 0.f,  0.f,  0.f};
  }
  // layer 0: 16 out, K=12 (padded 9) -> 3 WMMAs
  v8f c = P.cb0;
#pragma unroll
  for (int s = 0; s < 3; ++s) {
    v2f b = *(const v2f*)(actT + n * 16 + 4 * s + kh);
    c = wmma4(P.aw0[s], b, c);
  }
#pragma unroll
  for (int r = 0; r < 8; ++r) c[r] = fast_tanh(c[r]);
  *(v4f*)(actT + n * 16 + 8 * h + 0) = (v4f){c[0], c[1], c[2], c[3]};
  *(v4f*)(actT + n * 16 + 8 * h + 4) = (v4f){c[4], c[5], c[6], c[7]};

  // layer 1: 16 out, K=16 -> 4 WMMAs
  c = P.cb1;
#pragma unroll
  for (int s = 0; s < 4; ++s) {
    v2f b = *(const v2f*)(actT + n * 16 + 4 * s + kh);
    c = wmma4(P.aw1[s], b, c);
  }
#pragma unroll
  for (int r = 0; r < 8; ++r) c[r] = fast_tanh(c[r]);
  *(v4f*)(actT + n * 16 + 8 * h + 0) = (v4f){c[0], c[1], c[2], c[3]};
  *(v4f*)(actT + n * 16 + 8 * h + 4) = (v4f){c[4], c[5], c[6], c[7]};

  // layer 2: 8 out (rows 8..15 zero-padded), K=16 -> 4 WMMAs
  c = P.cb2;
#pragma unroll
  for (int s = 0; s < 4; ++s) {
    v2f b = *(const v2f*)(actT + n * 16 + 4 * s + kh);
    c = wmma4(P.aw2[s], b, c);
  }
#pragma unroll
  for (int r = 0; r < 8; ++r) c[r] = P.scale * fast_tanh(fast_tanh(c[r]));
  return c;  // rows 0..7 live on lanes 0..15; upper half is exactly zero
}

extern "C" __global__ __launch_bounds__(32) void ode_rnn_cdna5(
    const float* __restrict__ ts, const float* __restrict__ obs,
    const float* __restrict__ scale_p,
    const float* __restrict__ w0, const float* __restrict__ b0,
    const float* __restrict__ w1, const float* __restrict__ b1,
    const float* __restrict__ w2, const float* __restrict__ b2,
    const float* __restrict__ Wh, const float* __restrict__ Wx,
    const float* __restrict__ bx,
    const float* __restrict__ ow0, const float* __restrict__ ob0,
    const float* __restrict__ ow1, const float* __restrict__ ob1,
    const float* __restrict__ ow2, const float* __restrict__ ob2,
    float* __restrict__ out) {
  __shared__ float actT[16 * 16];   // transposed activation tile [col][k]
  __shared__ float obsT[16 * 32];   // observation tile [col][feature]

  const int  lane = threadIdx.x;
  const int  n    = lane & 15;      // batch column within tile
  const int  h    = lane >> 4;      // lane half
  const int  kh   = 2 * h;          // K offset contributed by lane half
  const bool lo   = (lane < 16);
  const int  seq  = blockIdx.x * TILE + n;

  // ---- preload recurrent weights into WMMA A-slices / bias accumulators ----
  VFW P;
#pragma unroll
  for (int s = 0; s < 3; ++s) P.aw0[s] = aslice(w0, 16, 9, n, 4 * s + kh);
#pragma unroll
  for (int s = 0; s < 4; ++s) P.aw1[s] = aslice(w1, 16, 16, n, 4 * s + kh);
#pragma unroll
  for (int s = 0; s < 4; ++s) P.aw2[s] = aslice(w2, 8, 16, n, 4 * s + kh);
  P.cb0 = cbias(b0, 16, h);
  P.cb1 = cbias(b1, 16, h);
  P.cb2 = cbias(b2, 8, h);
  P.scale = scale_p[0];

  v2f aWh[2], aWx[8];
#pragma unroll
  for (int s = 0; s < 2; ++s) aWh[s] = aslice(Wh, 8, 8, n, 4 * s + kh);
#pragma unroll
  for (int s = 0; s < 8; ++s) aWx[s] = aslice(Wx, 8, 32, n, 4 * s + kh);
  const v8f cbx = cbias(bx, 8, h);

  v8f y;
#pragma unroll
  for (int r = 0; r < 8; ++r) y[r] = 0.0f;
  float t_prev = ts[(size_t)seq * Nseq + 0];

  // ======================= sequential recurrence =======================
#pragma unroll 1
  for (int step = 0; step < Nseq; ++step) {
    // Issue this step's observation loads now; they are consumed only after
    // the 8 ODE sub-steps (plenty of WMMA work to hide VMEM latency).
    const float* op = obs + ((size_t)seq * Nseq + step) * Iin + 16 * h;
    v4f ob[4];
#pragma unroll
    for (int q = 0; q < 4; ++q) ob[q] = *(const v4f*)(op + 4 * q);
    if (step + 1 < Nseq) __builtin_prefetch(op + Iin, 0, 0);  // global_prefetch_b8

    const float t1 = ts[(size_t)seq * Nseq + step];
    const float dt = (t1 - t_prev) * (1.0f / NSUB);

#pragma unroll 1
    for (int sub = 0; sub < NSUB; ++sub) {
      const float t0 = t_prev + sub * dt;
      v8f yt;

      v8f k1 = vf_eval(P, actT, n, kh, h, lo, t0, y);
#pragma unroll
      for (int r = 0; r < 8; ++r) yt[r] = y[r] + dt * (A21f * k1[r]);
      v8f k2 = vf_eval(P, actT, n, kh, h, lo, t0 + C2f * dt, yt);
#pragma unroll
      for (int r = 0; r < 8; ++r) yt[r] = y[r] + dt * (A31f * k1[r] + A32f * k2[r]);
      v8f k3 = vf_eval(P, actT, n, kh, h, lo, t0 + C3f * dt, yt);
#pragma unroll
      for (int r = 0; r < 8; ++r)
        yt[r] = y[r] + dt * (A41f * k1[r] + A42f * k2[r] + A43f * k3[r]);
      v8f k4 = vf_eval(P, actT, n, kh, h, lo, t0 + C4f * dt, yt);
#pragma unroll
      for (int r = 0; r < 8; ++r)
        yt[r] = y[r] + dt * (A51f * k1[r] + A52f * k2[r] + A53f * k3[r] + A54f * k4[r]);
      v8f k5 = vf_eval(P, actT, n, kh, h, lo, t0 + C5f * dt, yt);
#pragma unroll
      for (int r = 0; r < 8; ++r)
        yt[r] = y[r] + dt * (A61f * k1[r] + A62f * k2[r] + A63f * k3[r] +
                             A64f * k4[r] + A65f * k5[r]);
      v8f k6 = vf_eval(P, actT, n, kh, h, lo, t0 + dt, yt);
#pragma unroll
      for (int r = 0; r < 8; ++r)
        y[r] += dt * (B1f * k1[r] + B2f * k2[r] + B3f * k3[r] +
                      B4f * k4[r] + B5f * k5[r] + B6f * k6[r]);
    }

    // ---------------- RNN cell: h = tanh(Wh h' + Wx x + bx) ----------------
    if (lo) {  // stage h' (K rows 0..7) into the transposed tile
      *(v4f*)(actT + n * 16 + 0) = (v4f){y[0], y[1], y[2], y[3]};
      *(v4f*)(actT + n * 16 + 4) = (v4f){y[4], y[5], y[6], y[7]};
    }
#pragma unroll
    for (int q = 0; q < 4; ++q)  // stage observation tile [col][feature]
      *(v4f*)(obsT + n * 32 + 16 * h + 4 * q) = ob[q];

    v8f c = cbx;
#pragma unroll
    for (int s = 0; s < 2; ++s) {  // Wh h' : K=8
      v2f b = *(const v2f*)(actT + n * 16 + 4 * s + kh);
      c = wmma4(aWh[s], b, c);
    }
#pragma unroll
    for (int s = 0; s < 8; ++s) {  // Wx x : K=32
      v2f b = *(const v2f*)(obsT + n * 32 + 4 * s + kh);
      c = wmma4(aWx[s], b, c);
    }
#pragma unroll
    for (int r = 0; r < 8; ++r) y[r] = fast_tanh(c[r]);

    t_prev = t1;
  }

  // ======================= output MLP (once) =======================
  v2f ao0[2], ao1[4], ao2[4];
#pragma unroll
  for (int s = 0; s < 2; ++s) ao0[s] = aslice(ow0, 16, 8, n, 4 * s + kh);
#pragma unroll
  for (int s = 0; s < 4; ++s) ao1[s] = aslice(ow1, 16, 16, n, 4 * s + kh);
#pragma unroll
  for (int s = 0; s < 4; ++s) ao2[s] = aslice(ow2, 8, 16, n, 4 * s + kh);
  const v8f co0 = cbias(ob0, 16, h), co1 = cbias(ob1, 16, h), co2 = cbias(ob2, 8, h);

  if (lo) {
    *(v4f*)(actT + n * 16 + 0) = (v4f){y[0], y[1], y[2], y[3]};
    *(v4f*)(actT + n * 16 + 4) = (v4f){y[4], y[5], y[6], y[7]};
  }
  v8f c = co0;
#pragma unroll
  for (int s = 0; s < 2; ++s)
    c = wmma4(ao0[s], *(const v2f*)(actT + n * 16 + 4 * s + kh), c);
#pragma unroll
  for (int r = 0; r < 8; ++r) c[r] = fmaxf(c[r], 0.0f);
  *(v4f*)(actT + n * 16 + 8 * h + 0) = (v4f){c[0], c[1], c[2], c[3]};
  *(v4f*)(actT + n * 16 + 8 * h + 4) = (v4f){c[4], c[5], c[6], c[7]};

  c = co1;
#pragma unroll
  for (int s = 0; s < 4; ++s)
    c = wmma4(ao1[s], *(const v2f*)(actT + n * 16 + 4 * s + kh), c);
#pragma unroll
  for (int r = 0; r < 8; ++r) c[r] = fmaxf(c[r], 0.0f);
  *(v4f*)(actT + n * 16 + 8 * h + 0) = (v4f){c[0], c[1], c[2], c[3]};
  *(v4f*)(actT + n * 16 + 8 * h + 4) = (v4f){c[4], c[5], c[6], c[7]};

  c = co2;
#pragma unroll
  for (int s = 0; s < 4; ++s)
    c = wmma4(ao2[s], *(const v2f*)(actT + n * 16 + 4 * s + kh), c);

  if (lo) {  // tuple output: [B,O] last outputs, then [B,H] final hidden
    float* po = out + (size_t)seq * Od;
    *(v4f*)(po + 0) = (v4f){c[0], c[1], c[2], c[3]};
    *(v4f*)(po + 4) = (v4f){c[4], c[5], c[6], c[7]};
    float* ph = out + (size_t)Bsz * Od + (size_t)seq * Hd;
    *(v4f*)(ph + 0) = (v4f){y[0], y[1], y[2], y[3]};
    *(v4f*)(ph + 4) = (v4f){y[4], y[5], y[6], y[7]};
  }
}

extern "C" void kernel_launch(void* const* d_in, const int* in_sizes, int n_in,
                              void* d_out, int out_size, void* d_ws, size_t ws_size,
                              hipStream_t stream) {
  (void)in_sizes; (void)n_in; (void)d_ws; (void)ws_size; (void)out_size;
  const float* ts   = (const float*)d_in[0];
  const float* obs  = (const float*)d_in[1];
  const float* sc   = (const float*)d_in[2];
  const float* w0   = (const float*)d_in[3];
  const float* b0   = (const float*)d_in[4];
  const float* w1   = (const float*)d_in[5];
  const float* b1   = (const float*)d_in[6];
  const float* w2   = (const float*)d_in[7];
  const float* b2   = (const float*)d_in[8];
  const float* Wh   = (const float*)d_in[9];
  const float* Wx   = (const float*)d_in[10];
  const float* bx   = (const float*)d_in[11];
  const float* ow0  = (const float*)d_in[12];
  const float* ob0  = (const float*)d_in[13];
  const float* ow1  = (const float*)d_in[14];
  const float* ob1  = (const float*)d_in[15];
  const float* ow2  = (const float*)d_in[16];
  const float* ob2  = (const float*)d_in[17];
  float* out = (float*)d_out;

  ode_rnn_cdna5<<<dim3(Bsz / TILE), dim3(32), 0, stream>>>(
      ts, obs, sc, w0, b0, w1, b1, w2, b2, Wh, Wx, bx,
      ow0, ob0, ow1, ob1, ow2, ob2, out);
}